// Resample1d_77970836291697
// MI455X (gfx1250) — compile-verified
//
#include <hip/hip_runtime.h>
#include <hip/hip_bf16.h>
#include <math.h>
#include <stdint.h>

// Fixed problem shape from the reference: input1 (4,64,256,512) f32,
// input2 (4,1,256,512) f32, output (4,64,256,512) f32.
#define B_  4
#define C_  64
#define H_  256
#define W_  512
#define THREADS 256
#define CPB 32                        // channels per block
#define NBLK (B_ * H_ * (C_ / CPB))   // 2048 blocks

typedef float v2f __attribute__((ext_vector_type(2)));
typedef int   v2i __attribute__((ext_vector_type(2)));

#if defined(__HIP_DEVICE_COMPILE__)
#  if __has_builtin(__builtin_amdgcn_global_load_async_to_lds_b64)
#    define USE_ASYNC 1
#  else
#    define USE_ASYNC 0
#  endif
#  if __has_builtin(__builtin_amdgcn_s_wait_asynccnt)
#    define WAIT_ASYNC(n) __builtin_amdgcn_s_wait_asynccnt(n)
#  else
#    define WAIT_ASYNC(n) asm volatile("s_wait_asynccnt %0" ::"i"(n) : "memory")
#  endif
#else
#  define USE_ASYNC 0
#  define WAIT_ASYNC(n)
#endif

#if USE_ASYNC
// Builtin signature (from hipcc diagnostic): first param is an AS1 pointer to
// a 2 x i32 vector (the 8-byte payload), second is the AS3 (LDS) counterpart.
typedef __attribute__((address_space(1))) v2i as1_v2i;
typedef __attribute__((address_space(3))) v2i as3_v2i;
#endif

// Stage one 512-float input row (2 KiB) into LDS: 256 threads x 8 bytes.
__device__ __forceinline__ void prefetch_row(const float* __restrict__ grow,
                                             float* lrow, int tid) {
#if USE_ASYNC
  // gfx1250 async global -> LDS copy, tracked by ASYNCcnt.
  __builtin_amdgcn_global_load_async_to_lds_b64(
      (as1_v2i*)(grow + 2 * tid), (as3_v2i*)(lrow + 2 * tid),
      /*offset=*/0, /*cpol=*/0);
#else
  *(v2f*)(lrow + 2 * tid) = *(const v2f*)(grow + 2 * tid);
#endif
}

__global__ __launch_bounds__(THREADS) void resample1d_kernel(
    const float* __restrict__ in1, const float* __restrict__ disp,
    float* __restrict__ out) {
  __shared__ float rowbuf[2][W_];

  const int tid   = threadIdx.x;
  const int blk   = blockIdx.x;
  const int bh    = blk / (C_ / CPB);          // b*H + h
  const int cbase = (blk % (C_ / CPB)) * CPB;  // first channel of this block
  const int b     = bh / H_;
  const int h     = bh % H_;

  const size_t chan_stride = (size_t)H_ * W_;
  const float* in_base  = in1 + ((size_t)b * C_ * H_ + h) * (size_t)W_;
  float*       out_base = out + ((size_t)b * C_ * H_ + h) * (size_t)W_;
  const float* drow     = disp + (size_t)bh * W_;

  // Kick off the first row fetch while we compute the resample coefficients.
  prefetch_row(in_base + (size_t)cbase * chan_stride, rowbuf[0], tid);

  // Per-thread coefficients for w = 2*tid and w = 2*tid+1. These depend only
  // on (b,h,w), so they live in registers and are reused for all CPB channels.
  const int wA = 2 * tid;
  const v2f dv = *(const v2f*)(drow + wA);

  float xA  = (float)wA + dv.x;
  float x0A = floorf(xA);
  float fA  = xA - x0A;
  float vA  = (xA >= 0.0f && xA <= (float)(W_ - 1)) ? 1.0f : 0.0f;
  int iA0 = (int)x0A;
  iA0 = iA0 < 0 ? 0 : (iA0 > W_ - 1 ? W_ - 1 : iA0);
  int iA1 = (iA0 + 1 > W_ - 1) ? W_ - 1 : iA0 + 1;
  const float cA0 = (1.0f - fA) * vA;
  const float cA1 = fA * vA;

  float xB  = (float)(wA + 1) + dv.y;
  float x0B = floorf(xB);
  float fB  = xB - x0B;
  float vB  = (xB >= 0.0f && xB <= (float)(W_ - 1)) ? 1.0f : 0.0f;
  int iB0 = (int)x0B;
  iB0 = iB0 < 0 ? 0 : (iB0 > W_ - 1 ? W_ - 1 : iB0);
  int iB1 = (iB0 + 1 > W_ - 1) ? W_ - 1 : iB0 + 1;
  const float cB0 = (1.0f - fB) * vB;
  const float cB1 = fB * vB;

  // Double-buffered channel loop: prefetch row c+1 while computing row c.
  for (int k = 0; k < CPB; ++k) {
    const int c = cbase + k;
    if (k + 1 < CPB)
      prefetch_row(in_base + (size_t)(c + 1) * chan_stride,
                   rowbuf[(k + 1) & 1], tid);
#if USE_ASYNC
    if (k + 1 < CPB) {
      WAIT_ASYNC(1);  // oldest (row c) done; row c+1 still in flight
    } else {
      WAIT_ASYNC(0);
    }
#endif
    __syncthreads();  // row c visible to all waves

    const float* rb = rowbuf[k & 1];
    v2f r;
    r.x = cA0 * rb[iA0] + cA1 * rb[iA1];
    r.y = cB0 * rb[iB0] + cB1 * rb[iB1];
    // Streaming output (written once, never re-read): non-temporal store.
    __builtin_nontemporal_store(r, (v2f*)(out_base + (size_t)c * chan_stride + wA));

    __syncthreads();  // everyone done reading rb before it is overwritten
  }
}

extern "C" void kernel_launch(void* const* d_in, const int* in_sizes, int n_in,
                              void* d_out, int out_size, void* d_ws,
                              size_t ws_size, hipStream_t stream) {
  (void)in_sizes; (void)n_in; (void)d_ws; (void)ws_size; (void)out_size;
  const float* in1  = (const float*)d_in[0];
  const float* disp = (const float*)d_in[1];
  float* out        = (float*)d_out;
  hipLaunchKernelGGL(resample1d_kernel, dim3(NBLK), dim3(THREADS), 0, stream,
                     in1, disp, out);
}